// SumOfTiledHyperCubeBasisFcns_68143951118418
// MI455X (gfx1250) — compile-verified
//
#include <hip/hip_runtime.h>

// ---------------------------------------------------------------------------
// SumOfTiledHyperCubeBasisFcns on MI455X (gfx1250, wave32)
//
// Phase 1 (negligible, 64^3 = 1MB): build 3-D inclusive prefix-sum with the
//   f32 WMMA pipe: cumsum == matmul with triangular ones matrix.
//     cumsum along contiguous axis:  C(16x64 strip) = V(16x64) * U(64x64)
//     cumsum along strided axes:     D = L(64x64)   * M(64xNcols)
//   One wave per 16x16 output tile, V_WMMA_F32_16X16X4_F32, 16 k-steps,
//   uniform break over all-zero triangular tiles. EXEC all-ones (32-thread
//   blocks, no divergence around WMMA).
// Phase 2: pad to 65^3 SAT P (P[a,b,c] = 0 if any index == 0).
// Phase 3 (dominant, memory-bound): per sample compute first/last indices
//   exactly as the reference (divide by step = 1.0f/63.0f, ceil/floor, clamp
//   to [0,63]) and 8-corner inclusion-exclusion gather from P.
//   4 samples/lane -> 3x global_load_b128 in, 1x global_store_b128 out,
//   128 MB streamed @ 23.3 TB/s ~= 5.5 us; SAT gathers are L2/WGP$-resident.
// ---------------------------------------------------------------------------

typedef float v2f  __attribute__((ext_vector_type(2)));
typedef float v4f  __attribute__((ext_vector_type(4)));
typedef float v8f  __attribute__((ext_vector_type(8)));

#define NC 64
#define NP 65
#define NSMPS 8388608

// ---------------------------------------------------------------------------
// Kernel 1: C = V * U  (inclusive cumsum along the contiguous axis).
// V: 4096x64 (lines as rows, contiguous), U[k][c] = (k <= c).
// One wave handles a 16-row strip: 4 col tiles x up to 16 k-steps.
// ---------------------------------------------------------------------------
__global__ __launch_bounds__(32) void scan_rows_VU(const float* __restrict__ src,
                                                   float* __restrict__ dst) {
  const int wave = blockIdx.x;        // 0..255 (4096 rows / 16)
  const int lane = threadIdx.x;       // 0..31
  const int m    = lane & 15;         // row within 16x16 tile / col of B
  const int half = lane >> 4;         // 0: K{0,1}, 1: K{2,3}
  const int rowBase = wave * 16;
  const float* rowPtr = src + (size_t)(rowBase + m) * NC;

  for (int n = 0; n < 4; ++n) {       // output col tile
    v8f acc = {0.f, 0.f, 0.f, 0.f, 0.f, 0.f, 0.f, 0.f};
    const int cmax = n * 16 + 15;
    for (int kt = 0; kt < 16; ++kt) { // K = 4 per step
      const int r0 = kt * 4 + half * 2;   // absolute K index of lane's pair
      if (kt * 4 > cmax) break;           // U tile all-zero (uniform branch)
      // A tile: lane holds V[m][r0], V[m][r0+1]  (adjacent -> b64 load)
      v2f a;
      a[0] = rowPtr[r0];
      a[1] = rowPtr[r0 + 1];
      // B tile: lane holds U[r0][col], U[r0+1][col], col = n*16 + m
      const int col = n * 16 + m;
      v2f b;
      b[0] = (r0     <= col) ? 1.0f : 0.0f;
      b[1] = (r0 + 1 <= col) ? 1.0f : 0.0f;
      acc = __builtin_amdgcn_wmma_f32_16x16x4_f32(false, a, false, b,
                                                  (short)0, acc, false, false);
    }
    // D: VGPR v -> row = v + 8*half, col = m (within tile)
    #pragma unroll
    for (int v = 0; v < 8; ++v)
      dst[(size_t)(rowBase + v + 8 * half) * NC + n * 16 + m] = acc[v];
  }
}

// ---------------------------------------------------------------------------
// Kernel 2: D = L * M  (inclusive cumsum along rows of M), L[a][b] = (b <= a).
// M: 64 rows x (colTiles*16) cols, row stride rowStride, per-slab.
// One wave per 16x16 output tile.
// ---------------------------------------------------------------------------
__global__ __launch_bounds__(32) void scan_cols_LM(const float* __restrict__ src,
                                                   float* __restrict__ dst,
                                                   int rowStride, int slabStride,
                                                   int colTiles) {
  const int id   = blockIdx.x;
  const int ct   = id % colTiles;
  const int mt   = (id / colTiles) & 3;
  const int slab = id / (colTiles * 4);
  const int lane = threadIdx.x;
  const int m    = lane & 15;
  const int half = lane >> 4;

  const float* S = src + (size_t)slab * slabStride;
  float*       D = dst + (size_t)slab * slabStride;
  const int col  = ct * 16 + m;
  const int rmax = mt * 16 + 15;

  v8f acc = {0.f, 0.f, 0.f, 0.f, 0.f, 0.f, 0.f, 0.f};
  for (int kt = 0; kt < 16; ++kt) {
    if (kt * 4 > rmax) break;             // L tile all-zero (uniform branch)
    const int r0   = kt * 4 + half * 2;   // absolute K index of lane's pair
    const int rowm = mt * 16 + m;
    // A tile: A[m][kk] = L[rowm][kt*4+kk] = (kt*4+kk <= rowm)
    v2f a;
    a[0] = (r0     <= rowm) ? 1.0f : 0.0f;
    a[1] = (r0 + 1 <= rowm) ? 1.0f : 0.0f;
    // B tile: lane holds M[r0][col], M[r0+1][col]
    v2f b;
    b[0] = S[(size_t)r0 * rowStride + col];
    b[1] = S[(size_t)(r0 + 1) * rowStride + col];
    acc = __builtin_amdgcn_wmma_f32_16x16x4_f32(false, a, false, b,
                                                (short)0, acc, false, false);
  }
  #pragma unroll
  for (int v = 0; v < 8; ++v)
    D[(size_t)(mt * 16 + v + 8 * half) * rowStride + col] = acc[v];
}

// ---------------------------------------------------------------------------
// Kernel 3: pad triple-cumsum S (64^3) into SAT P (65^3), zeros at index 0.
// ---------------------------------------------------------------------------
__global__ __launch_bounds__(256) void pad_kernel(const float* __restrict__ S,
                                                  float* __restrict__ P) {
  const int t = blockIdx.x * blockDim.x + threadIdx.x;
  const int total = NP * NP * NP;
  if (t >= total) return;
  const int c = t % NP;
  const int b = (t / NP) % NP;
  const int a = t / (NP * NP);
  float v = 0.0f;
  if (a && b && c)
    v = S[((size_t)(a - 1) * NC + (b - 1)) * NC + (c - 1)];
  P[t] = v;
}

// ---------------------------------------------------------------------------
// Kernel 4: dominant gather phase. 4 samples per lane.
// ---------------------------------------------------------------------------
__device__ __forceinline__ void active_bounds(float xd, int& lo, int& hi) {
  // Exactly mirror the reference: step = float32(1/63), divide, ceil/floor,
  // clip to [0, 63], cast to int.
  const float step = 1.0f / 63.0f;
  float f = ceilf((xd - 0.05f) / step);
  float l = floorf((xd + 0.05f) / step);
  f = fminf(fmaxf(f, 0.0f), 63.0f);
  l = fminf(fmaxf(l, 0.0f), 63.0f);
  lo = (int)f;
  hi = (int)l;
}

__device__ __forceinline__ float corner_sum(const float* __restrict__ P,
                                            float x0, float x1, float x2) {
  int a0, a1, b0, b1, c0, c1;
  active_bounds(x0, a0, a1);
  active_bounds(x1, b0, b1);
  active_bounds(x2, c0, c1);
  // P index (a*65 + b)*65 + c ; signs: (-1)^(#first-corners)
  const int A0 = a0 * (NP * NP), A1 = a1 * (NP * NP);
  const int B0 = b0 * NP,        B1 = b1 * NP;
  float r = P[A1 + B1 + c1]
          - P[A0 + B1 + c1] - P[A1 + B0 + c1] - P[A1 + B1 + c0]
          + P[A0 + B0 + c1] + P[A0 + B1 + c0] + P[A1 + B0 + c0]
          - P[A0 + B0 + c0];
  return r;
}

__global__ __launch_bounds__(256) void gather_kernel(const float* __restrict__ x,
                                                     const float* __restrict__ P,
                                                     float* __restrict__ y,
                                                     int nQuads) {
  const int t = blockIdx.x * blockDim.x + threadIdx.x;
  if (t >= nQuads) return;
  // 4 samples = 12 floats = 3 x b128 loads
  const v4f* xv = (const v4f*)x;
  v4f p0 = xv[3 * (size_t)t + 0];
  v4f p1 = xv[3 * (size_t)t + 1];
  v4f p2 = xv[3 * (size_t)t + 2];

  v4f out;
  out[0] = corner_sum(P, p0[0], p0[1], p0[2]);
  out[1] = corner_sum(P, p0[3], p1[0], p1[1]);
  out[2] = corner_sum(P, p1[2], p1[3], p2[0]);
  out[3] = corner_sum(P, p2[1], p2[2], p2[3]);

  ((v4f*)y)[t] = out;   // b128 store
}

// ---------------------------------------------------------------------------
extern "C" void kernel_launch(void* const* d_in, const int* in_sizes, int n_in,
                              void* d_out, int out_size, void* d_ws, size_t ws_size,
                              hipStream_t stream) {
  const float* x  = (const float*)d_in[0];   // (8388608, 3) f32
  const float* bm = (const float*)d_in[1];   // (64, 64, 64) f32
  float* y = (float*)d_out;                  // 8388608 f32

  float* buf0 = (float*)d_ws;                // 64^3
  float* buf1 = buf0 + (size_t)NC * NC * NC; // 64^3
  float* P    = buf1 + (size_t)NC * NC * NC; // 65^3

  // cumsum along k (contiguous): C = V * U, 4096 lines / 16 per wave
  scan_rows_VU<<<256, 32, 0, stream>>>(bm, buf0);
  // cumsum along j: per-slab D = L * M (64 slabs, rowStride 64, 4 col tiles)
  scan_cols_LM<<<64 * 4 * 4, 32, 0, stream>>>(buf0, buf1, NC, NC * NC, 4);
  // cumsum along i: D = L * M (1 slab, rowStride 4096, 256 col tiles)
  scan_cols_LM<<<4 * 256, 32, 0, stream>>>(buf1, buf0, NC * NC, 0, 256);
  // pad to 65^3 SAT
  const int totalP = NP * NP * NP;
  pad_kernel<<<(totalP + 255) / 256, 256, 0, stream>>>(buf0, P);
  // dominant gather phase: 4 samples per lane
  const int nQuads = NSMPS / 4;
  gather_kernel<<<(nQuads + 255) / 256, 256, 0, stream>>>(x, P, y, nQuads);
}